// RNN_27462020891075
// MI455X (gfx1250) — compile-verified
//
#include <hip/hip_runtime.h>
#include <hip/hip_bf16.h>

#define SEQ 131072
#define INPUT 256
#define HIDDEN 128
#define CHUNK 256
#define NCHUNK (SEQ / CHUNK) // 512
#define LOGS 264             // padded logits row stride (bank-conflict avoidance)
#define TBS 132              // padded Tbuf row stride

typedef __attribute__((ext_vector_type(2))) float f32x2;
typedef __attribute__((ext_vector_type(8))) float f32x8;

__device__ __forceinline__ f32x8 wmma_f32(f32x2 a, f32x2 b, f32x8 c) {
    // D = A(16x4) * B(4x16) + C(16x16), fp32 WMMA
    return __builtin_amdgcn_wmma_f32_16x16x4_f32(false, a, false, b, (short)0, c,
                                                 false, false);
}

// ---------------------------------------------------------------------------
// K1: U[i][h] = sum_k s[i][k] * Wx[h][k]   ([SEQ,256] x [128,256]^T -> [SEQ,128])
// One block per 16 sequence rows; wave w handles hidden tile h0 = 16*w.
// ---------------------------------------------------------------------------
__global__ __launch_bounds__(256) void k_gemm_u(const float* __restrict__ s,
                                                const float* __restrict__ Wx,
                                                float* __restrict__ U) {
    const int i0   = blockIdx.x * 16;
    const int wave = threadIdx.x >> 5;
    const int lane = threadIdx.x & 31;
    const int mn   = lane & 15;        // M for A-lanes, N for B/C-lanes
    const int half = lane >> 4;        // 0: K pair {0,1}, 1: K pair {2,3}
    const int koff = half * 2;
    const int h0   = wave * 16;

    const float* arow = s  + (size_t)(i0 + mn) * INPUT + koff;
    const float* brow = Wx + (size_t)(h0 + mn) * INPUT + koff;

    f32x8 acc = {};
    for (int k0 = 0; k0 < INPUT; k0 += 4) {
        f32x2 a = *(const f32x2*)(arow + k0);
        f32x2 b = *(const f32x2*)(brow + k0);
        acc = wmma_f32(a, b, acc);
    }
    float* Urow = U + (size_t)(i0 + half * 8) * HIDDEN + h0 + mn;
#pragma unroll
    for (int r = 0; r < 8; ++r) Urow[(size_t)r * HIDDEN] = acc[r];
}

// ---------------------------------------------------------------------------
// K2: dst = src * src  (128x128 matrix squaring via WMMA from LDS). 1 block.
// Launched 8 times -> Wh^256.
// ---------------------------------------------------------------------------
__global__ __launch_bounds__(256) void k_square(const float* __restrict__ src,
                                                float* __restrict__ dst) {
    __shared__ float A[HIDDEN * HIDDEN]; // 64 KB
    for (int i = threadIdx.x; i < HIDDEN * HIDDEN; i += 256) A[i] = src[i];
    __syncthreads();

    const int wave = threadIdx.x >> 5, lane = threadIdx.x & 31;
    const int mn = lane & 15, half = lane >> 4, koff = half * 2;

    for (int t = wave; t < 64; t += 8) { // 8x8 grid of 16x16 output tiles
        const int m0 = (t >> 3) * 16, n0 = (t & 7) * 16;
        f32x8 acc = {};
        for (int k0 = 0; k0 < HIDDEN; k0 += 4) {
            const int kb = k0 + koff;
            f32x2 a = *(const f32x2*)&A[(m0 + mn) * HIDDEN + kb];
            f32x2 b;
            b.x = A[kb * HIDDEN + n0 + mn];
            b.y = A[(kb + 1) * HIDDEN + n0 + mn];
            acc = wmma_f32(a, b, acc);
        }
#pragma unroll
        for (int r = 0; r < 8; ++r)
            dst[(size_t)(m0 + half * 8 + r) * HIDDEN + n0 + mn] = acc[r];
    }
}

// ---------------------------------------------------------------------------
// K3: per-chunk zero-entry effect b_c: t=0; for j in chunk: t = Wh t + U_j.
// Wh stored transposed in LDS -> conflict-free lane-parallel matvec.
// ---------------------------------------------------------------------------
__global__ __launch_bounds__(128) void k_chunk_b(const float* __restrict__ Wh,
                                                 const float* __restrict__ U,
                                                 float* __restrict__ bvec) {
    __shared__ float WhT[HIDDEN * HIDDEN]; // 64 KB, WhT[k][h] = Wh[h][k]
    __shared__ float tb[2][HIDDEN];
    const int tid = threadIdx.x;
    const int c   = blockIdx.x;

    for (int r = 0; r < HIDDEN; ++r) WhT[r * HIDDEN + tid] = Wh[tid * HIDDEN + r];
    tb[0][tid] = 0.0f;
    __syncthreads();

    const float* Uc = U + (size_t)c * CHUNK * HIDDEN;
    int cur = 0;
    for (int j = 0; j < CHUNK; ++j) {
        float acc = Uc[(size_t)j * HIDDEN + tid];
        const float* tc = tb[cur];
#pragma unroll 4
        for (int k = 0; k < HIDDEN; ++k) acc += WhT[k * HIDDEN + tid] * tc[k];
        tb[cur ^ 1][tid] = acc;
        __syncthreads();
        cur ^= 1;
    }
    bvec[(size_t)c * HIDDEN + tid] = tb[cur][tid];
}

// ---------------------------------------------------------------------------
// K4: chunk-level scan E_{c+1} = P*E_c + b_c, E_0 = initialState. Final -> d_out.
// ---------------------------------------------------------------------------
__global__ __launch_bounds__(128) void k_scan(const float* __restrict__ P,
                                              const float* __restrict__ bvec,
                                              const float* __restrict__ init,
                                              float* __restrict__ E,
                                              float* __restrict__ outFinal) {
    __shared__ float PT[HIDDEN * HIDDEN];
    __shared__ float eb[2][HIDDEN];
    const int tid = threadIdx.x;
    for (int r = 0; r < HIDDEN; ++r) PT[r * HIDDEN + tid] = P[tid * HIDDEN + r];
    eb[0][tid] = init[tid];
    __syncthreads();

    int cur = 0;
    for (int c = 0; c < NCHUNK; ++c) {
        E[(size_t)c * HIDDEN + tid] = eb[cur][tid];
        float acc = bvec[(size_t)c * HIDDEN + tid];
        const float* ec = eb[cur];
#pragma unroll 4
        for (int k = 0; k < HIDDEN; ++k) acc += PT[k * HIDDEN + tid] * ec[k];
        eb[cur ^ 1][tid] = acc;
        __syncthreads();
        cur ^= 1;
    }
    outFinal[tid] = eb[cur][tid]; // t_final
}

// ---------------------------------------------------------------------------
// K5: replay each chunk with the true entry state; every 16 steps compute
// logits = Tbuf * Wy^T with WMMA, then log-softmax + KLDiv partial loss.
// ---------------------------------------------------------------------------
__global__ __launch_bounds__(256) void k_chunk_loss(
    const float* __restrict__ Wh, const float* __restrict__ Wy,
    const float* __restrict__ U, const float* __restrict__ E,
    const float* __restrict__ label, float* __restrict__ partial) {
    __shared__ float WhT[HIDDEN * HIDDEN];  // 64 KB
    __shared__ float tb[2][HIDDEN];
    __shared__ float Tbuf[16 * TBS];        // 16 buffered states
    __shared__ float logits[16 * LOGS];     // 16 x 256 logits (padded)
    __shared__ float red[256];

    const int tid = threadIdx.x;
    const int c   = blockIdx.x;
    for (int i = tid; i < HIDDEN * HIDDEN; i += 256)
        WhT[i] = Wh[(i & (HIDDEN - 1)) * HIDDEN + (i >> 7)];
    if (tid < HIDDEN) tb[0][tid] = E[(size_t)c * HIDDEN + tid];
    __syncthreads();

    const float* Uc  = U + (size_t)c * CHUNK * HIDDEN;
    const int row    = tid >> 4;   // 0..15  (loss phase)
    const int sub    = tid & 15;   // 0..15
    const int wave   = tid >> 5, lane = tid & 31;
    const int mn     = lane & 15, half = lane >> 4, koff = half * 2;

    int   cur     = 0;
    float lossAcc = 0.0f;

    for (int g = 0; g < CHUNK / 16; ++g) {
        // --- 16 sequential recurrence steps (threads 0..127) ---
        for (int jj = 0; jj < 16; ++jj) {
            if (tid < HIDDEN) {
                const int j = g * 16 + jj;
                float acc = Uc[(size_t)j * HIDDEN + tid];
                const float* tc = tb[cur];
#pragma unroll 4
                for (int k = 0; k < HIDDEN; ++k) acc += WhT[k * HIDDEN + tid] * tc[k];
                tb[cur ^ 1][tid] = acc;
                Tbuf[jj * TBS + tid] = acc;
            }
            __syncthreads();
            cur ^= 1;
        }
        // --- logits[16x256] = Tbuf[16x128] * Wy^T via WMMA (all 8 waves) ---
        for (int t = wave; t < 16; t += 8) {
            const int c0 = t * 16;
            f32x8 acc = {};
            for (int k0 = 0; k0 < HIDDEN; k0 += 4) {
                const int kb = k0 + koff;
                f32x2 a = *(const f32x2*)&Tbuf[mn * TBS + kb];
                f32x2 b = *(const f32x2*)&Wy[(size_t)(c0 + mn) * HIDDEN + kb];
                acc = wmma_f32(a, b, acc);
            }
#pragma unroll
            for (int r = 0; r < 8; ++r)
                logits[(r + half * 8) * LOGS + c0 + mn] = acc[r];
        }
        __syncthreads();
        // --- log-softmax + KL for 16 rows (16 lanes per row) ---
        {
            const float* lrow = &logits[row * LOGS];
            float mx = -3.4e38f;
#pragma unroll
            for (int q = 0; q < 16; ++q) mx = fmaxf(mx, lrow[sub + q * 16]);
            for (int off = 8; off > 0; off >>= 1) mx = fmaxf(mx, __shfl_xor(mx, off, 16));
            float se = 0.0f;
#pragma unroll
            for (int q = 0; q < 16; ++q) se += expf(lrow[sub + q * 16] - mx);
            for (int off = 8; off > 0; off >>= 1) se += __shfl_xor(se, off, 16);
            const float lse = logf(se) + mx;
            const float* labrow = label + ((size_t)c * CHUNK + g * 16 + row) * INPUT;
            float pl = 0.0f;
#pragma unroll
            for (int q = 0; q < 16; ++q) {
                const int col = sub + q * 16;
                const float lb = labrow[col];
                pl += lb * (logf(lb) - (lrow[col] - lse));
            }
            lossAcc += pl;
        }
        __syncthreads();
    }
    // deterministic block reduction
    red[tid] = lossAcc;
    __syncthreads();
    for (int s2 = 128; s2 > 0; s2 >>= 1) {
        if (tid < s2) red[tid] += red[tid + s2];
        __syncthreads();
    }
    if (tid == 0) partial[c] = red[0];
}

// ---------------------------------------------------------------------------
// K6: fixed-order reduction of NCHUNK partials -> d_out[HIDDEN] (loss * 1/INPUT)
// ---------------------------------------------------------------------------
__global__ __launch_bounds__(256) void k_reduce(const float* __restrict__ partial,
                                                float* __restrict__ out) {
    __shared__ float red[256];
    const int tid = threadIdx.x;
    red[tid] = partial[tid] + partial[tid + 256];
    __syncthreads();
    for (int s = 128; s > 0; s >>= 1) {
        if (tid < s) red[tid] += red[tid + s];
        __syncthreads();
    }
    if (tid == 0) out[HIDDEN] = red[0] * (1.0f / INPUT);
}

extern "C" void kernel_launch(void* const* d_in, const int* in_sizes, int n_in,
                              void* d_out, int out_size, void* d_ws, size_t ws_size,
                              hipStream_t stream) {
    (void)in_sizes; (void)n_in; (void)out_size; (void)ws_size;
    const float* s     = (const float*)d_in[0];
    const float* label = (const float*)d_in[1];
    const float* init  = (const float*)d_in[2];
    const float* Wx    = (const float*)d_in[3];
    const float* Wh    = (const float*)d_in[4];
    const float* Wy    = (const float*)d_in[5];
    float* out = (float*)d_out;

    float* ws = (float*)d_ws;
    float* U  = ws;                           // SEQ*HIDDEN        (64 MB)
    float* Pa = U + (size_t)SEQ * HIDDEN;     // 128*128
    float* Pb = Pa + HIDDEN * HIDDEN;         // 128*128
    float* bv = Pb + HIDDEN * HIDDEN;         // NCHUNK*HIDDEN
    float* E  = bv + (size_t)NCHUNK * HIDDEN; // NCHUNK*HIDDEN
    float* pp = E + (size_t)NCHUNK * HIDDEN;  // NCHUNK

    k_gemm_u<<<SEQ / 16, 256, 0, stream>>>(s, Wx, U);

    // Wh^256 via 8 squarings (ping-pong in workspace)
    k_square<<<1, 256, 0, stream>>>(Wh, Pa);
    k_square<<<1, 256, 0, stream>>>(Pa, Pb);
    k_square<<<1, 256, 0, stream>>>(Pb, Pa);
    k_square<<<1, 256, 0, stream>>>(Pa, Pb);
    k_square<<<1, 256, 0, stream>>>(Pb, Pa);
    k_square<<<1, 256, 0, stream>>>(Pa, Pb);
    k_square<<<1, 256, 0, stream>>>(Pb, Pa);
    k_square<<<1, 256, 0, stream>>>(Pa, Pb); // Pb = Wh^256

    k_chunk_b<<<NCHUNK, 128, 0, stream>>>(Wh, U, bv);
    k_scan<<<1, 128, 0, stream>>>(Pb, bv, init, E, out);
    k_chunk_loss<<<NCHUNK, 256, 0, stream>>>(Wh, Wy, U, E, label, pp);
    k_reduce<<<1, 256, 0, stream>>>(pp, out);
}